// DenseGrid_87591563035291
// MI455X (gfx1250) — compile-verified
//
#include <hip/hip_runtime.h>

// DenseGrid trilinear sampling: out[n, c] = trilerp(grid[c], xyz[n])
// grid: (C=12, D=160, H=160, W=160) fp32, xyz: (N,3) fp32, out: (N,12) fp32.
//
// MI455X reasoning: pure random-gather workload. Grid = 196.6MB vs 192MB
// global L2 -> grid ~fits L2; win comes from (1) minimal gather count
// (compiler merges w-pairs into global_load_b64, 48 gathers/point),
// (2) deep MLP from fully unrolled independent loads, and (3) keeping the
// L2 dedicated to grid lines by marking all streaming traffic (xyz in,
// out writes) non-temporal (TH=NT) so it doesn't evict grid data.

#define C_CH 12
#define DIM_D 160
#define DIM_H 160
#define DIM_W 160
#define HW    (DIM_H * DIM_W)            // 25600
#define CS    (DIM_D * DIM_H * DIM_W)    // 4096000 elements per channel

// Native clang vector type: required by __builtin_nontemporal_store
// (HIP's float4 is a struct and is rejected).
typedef float v4f __attribute__((ext_vector_type(4)));

__device__ __forceinline__ void axis_coord(float p, float mn, float mx,
                                           float size_f, int& i0, float& f)
{
    // u = (p - mn) / (mx - mn) * (size-1), clamped to [0, size-1]
    float sc = (size_f - 1.0f) / (mx - mn);
    float u  = (p - mn) * sc;
    u = fminf(fmaxf(u, 0.0f), size_f - 1.0f);
    float fi = floorf(u);
    fi = fminf(fi, size_f - 2.0f);   // i0 in [0, size-2]
    f  = u - fi;                      // f in [0, 1]
    i0 = (int)fi;
}

__global__ __launch_bounds__(256) void
DenseGrid_trilerp_kernel(const float* __restrict__ xyz,
                         const float* __restrict__ grid,
                         const float* __restrict__ xyz_min,
                         const float* __restrict__ xyz_max,
                         float* __restrict__ out,
                         int n)
{
    const unsigned i = blockIdx.x * blockDim.x + threadIdx.x;
    if (i >= (unsigned)n) return;

    // Streaming, read-once point coordinates -> non-temporal (keep L2 for grid)
    const float* xp = xyz + (size_t)i * 3u;
    const float px = __builtin_nontemporal_load(xp + 0);
    const float py = __builtin_nontemporal_load(xp + 1);
    const float pz = __builtin_nontemporal_load(xp + 2);

    // Uniform bounds -> scalar loads
    const float mn0 = xyz_min[0], mn1 = xyz_min[1], mn2 = xyz_min[2];
    const float mx0 = xyz_max[0], mx1 = xyz_max[1], mx2 = xyz_max[2];

    int d0, h0, w0;
    float fd, fh, fw;
    axis_coord(px, mn0, mx0, (float)DIM_D, d0, fd);
    axis_coord(py, mn1, mx1, (float)DIM_H, h0, fh);
    axis_coord(pz, mn2, mx2, (float)DIM_W, w0, fw);

    // 8 trilinear weights
    const float gd0 = 1.0f - fd, gh0 = 1.0f - fh, gw0 = 1.0f - fw;
    const float w000 = gd0 * gh0 * gw0;
    const float w001 = gd0 * gh0 * fw;
    const float w010 = gd0 * fh  * gw0;
    const float w011 = gd0 * fh  * fw;
    const float w100 = fd  * gh0 * gw0;
    const float w101 = fd  * gh0 * fw;
    const float w110 = fd  * fh  * gw0;
    const float w111 = fd  * fh  * fw;

    // Base element index of corner (d0,h0,w0); all 8 corners are constant
    // immediate offsets from it: {0,1,W,W+1,HW,HW+1,HW+W,HW+W+1}.
    // w-pairs are adjacent -> compiler fuses into global_load_b64 (4/channel).
    const unsigned base = (unsigned)(d0 * HW + h0 * DIM_W + w0);

    float acc[C_CH];
    const float* gc = grid + (size_t)base;
#pragma unroll
    for (int c = 0; c < C_CH; ++c) {
        // 8 independent gathers with immediate offsets -> deep MLP batch.
        // Default TH=RT so grid lines stay resident in the 192MB L2.
        const float v000 = gc[0];
        const float v001 = gc[1];
        const float v010 = gc[DIM_W];
        const float v011 = gc[DIM_W + 1];
        const float v100 = gc[HW];
        const float v101 = gc[HW + 1];
        const float v110 = gc[HW + DIM_W];
        const float v111 = gc[HW + DIM_W + 1];

        float a;
        a = v000 * w000;
        a = fmaf(v001, w001, a);
        a = fmaf(v010, w010, a);
        a = fmaf(v011, w011, a);
        a = fmaf(v100, w100, a);
        a = fmaf(v101, w101, a);
        a = fmaf(v110, w110, a);
        a = fmaf(v111, w111, a);
        acc[c] = a;

        gc += (size_t)CS;   // next channel (16,384,000 B stride)
    }

    // out[i, 0..11]: 48B per point, 16B aligned -> three b128 NT stores.
    // Written once, never re-read -> non-temporal so they don't evict
    // grid lines from L2.
    v4f* o4 = (v4f*)(out + (size_t)i * C_CH);
    v4f t0 = { acc[0], acc[1], acc[2],  acc[3]  };
    v4f t1 = { acc[4], acc[5], acc[6],  acc[7]  };
    v4f t2 = { acc[8], acc[9], acc[10], acc[11] };
    __builtin_nontemporal_store(t0, o4 + 0);
    __builtin_nontemporal_store(t1, o4 + 1);
    __builtin_nontemporal_store(t2, o4 + 2);
}

extern "C" void kernel_launch(void* const* d_in, const int* in_sizes, int n_in,
                              void* d_out, int out_size, void* d_ws, size_t ws_size,
                              hipStream_t stream)
{
    const float* xyz     = (const float*)d_in[0];
    const float* grid    = (const float*)d_in[1];
    const float* xyz_min = (const float*)d_in[2];
    const float* xyz_max = (const float*)d_in[3];
    float* out           = (float*)d_out;

    const int n = in_sizes[0] / 3;   // N points
    const int block = 256;           // 8 wave32 waves
    const int gridsz = (n + block - 1) / block;

    DenseGrid_trilerp_kernel<<<gridsz, block, 0, stream>>>(
        xyz, grid, xyz_min, xyz_max, out, n);
}